// GIN_54211077210422
// MI455X (gfx1250) — compile-verified
//
#include <hip/hip_runtime.h>
#include <hip/hip_bf16.h>

// GIN forward on MI455X (gfx1250):
//   agg1 = segment_sum(feats[src], dst)          -> f32 hardware atomics (L2-resident)
//   h    = relu((feats+agg1) @ W1 + b1)          -> V_WMMA_F32_16X16X4_F32
//   agg2 = segment_sum(h[src], dst)              -> f32 hardware atomics
//   out  = segment_sum((h+agg2) @ W2 + b2, gid)  -> WMMA + fused pooled atomics
//
// All feature arrays (25.6 MB) are L2-resident (192 MB L2); the edge
// scatter/gather (~1.2 GB of L2 traffic, 153.6M f32 atomics) dominates.
// GEMMs are ~1.6 GFLOP total -> full f32 precision via WMMA 16x16x4.

typedef float v2f __attribute__((ext_vector_type(2)));
typedef float v8f __attribute__((ext_vector_type(8)));

#define D_FEAT 64

// ---------------------------------------------------------------------------
// Edge scatter-add: agg[dst[e]][0..63] += x[src[e]][0..63]
// 16 threads per edge, float4 (b128) per thread; atomics are hardware
// global_atomic_add_f32 (unavoidable minimum: one per edge-channel).
// ---------------------------------------------------------------------------
__global__ void __launch_bounds__(256)
gin_scatter_kernel(const float* __restrict__ x,
                   const int*   __restrict__ src,
                   const int*   __restrict__ dst,
                   float*       __restrict__ agg,
                   int nEdges)
{
    const int q = (threadIdx.x & 15) * 4;                 // channel quad
    const int e = blockIdx.x * 16 + (threadIdx.x >> 4);
    if (e >= nEdges) return;
    const int s = src[e];
    const int d = dst[e];
    const float4 v = *(const float4*)(x + (size_t)s * D_FEAT + q);
    float* o = agg + (size_t)d * D_FEAT + q;
    unsafeAtomicAdd(o + 0, v.x);
    unsafeAtomicAdd(o + 1, v.y);
    unsafeAtomicAdd(o + 2, v.z);
    unsafeAtomicAdd(o + 3, v.w);
}

// ---------------------------------------------------------------------------
// Fused (X + Xagg) @ W + bias, per-wave 16x64 output tile via WMMA f32 16x16x4.
//   mode 0: H[row][col] = relu(result)
//   mode 1: pool[gid[row]][col] += result    (fused SumPooling)
//
// WMMA VGPR layouts (CDNA5 ISA):
//   A (16x4 f32, 2 VGPRs):  lane m = lane&15 holds A[m][2*(lane>>4)+j] in VGPR j
//   B (4x16 f32, 2 VGPRs):  lane n = lane&15 holds B[2*(lane>>4)+j][n] in VGPR j
//   C/D (16x16 f32, 8 VGPRs): VGPR v, lanes 0-15 -> (M=v,   N=lane)
//                                     lanes16-31 -> (M=v+8, N=lane-16)
//
// W is staged in LDS pair-interleaved: sWp[(k>>1)*128 + col*2 + (k&1)]
// so each B fragment is a single aligned ds_load_b64.
// A rows are CLAMPED (not predicated): out-of-range rows only feed
// out-of-range output rows, which the guarded writeback drops. This keeps
// the inner loop free of EXEC divergence (WMMA requires EXEC all-ones).
// ---------------------------------------------------------------------------
__global__ void __launch_bounds__(256)
gin_layer_kernel(const float* __restrict__ X,
                 const float* __restrict__ Xagg,
                 const float* __restrict__ W,      // [64 in][64 out] row-major
                 const float* __restrict__ bias,   // [64]
                 float*       __restrict__ H,      // mode 0 output
                 const int*   __restrict__ gid,    // mode 1 graph ids
                 float*       __restrict__ pool,   // mode 1 output [nGraphs][64]
                 int nNodes, int mode)
{
    __shared__ float sWp[D_FEAT * D_FEAT];   // 16 KB, pair-interleaved
    __shared__ float sB[D_FEAT];

    for (int idx = threadIdx.x; idx < D_FEAT * D_FEAT; idx += 256) {
        const int k   = idx >> 6;
        const int col = idx & 63;
        sWp[(k >> 1) * 128 + col * 2 + (k & 1)] = W[idx];
    }
    if (threadIdx.x < D_FEAT) sB[threadIdx.x] = bias[threadIdx.x];
    __syncthreads();

    const int wave    = threadIdx.x >> 5;
    const int lane    = threadIdx.x & 31;
    const int rowTile = blockIdx.x * 8 + wave;         // one wave = 16 node rows
    const int rowBase = rowTile * 16;
    if (rowBase >= nNodes) return;

    const int halfSel = lane >> 4;                     // 0: lanes 0-15, 1: 16-31
    const int laneM   = lane & 15;

    v8f acc[4];
    #pragma unroll
    for (int ct = 0; ct < 4; ++ct) acc[ct] = (v8f){0.f,0.f,0.f,0.f,0.f,0.f,0.f,0.f};

    // Clamped A row: branch-free inner loop.
    int aRow = rowBase + laneM;
    if (aRow >= nNodes) aRow = nNodes - 1;
    const float* __restrict__ xr = X    + (size_t)aRow * D_FEAT;
    const float* __restrict__ gr = Xagg + (size_t)aRow * D_FEAT;

    #pragma unroll 4
    for (int kt = 0; kt < 16; ++kt) {                  // K = 64 in steps of 4
        const int k0 = kt * 4 + halfSel * 2;           // even -> 8B aligned
        const v2f ax = *(const v2f*)(xr + k0);
        const v2f ag = *(const v2f*)(gr + k0);
        const v2f a  = ax + ag;                        // v_pk_add_f32
        const int bRow = (kt * 2 + halfSel) * 128;
        #pragma unroll
        for (int ct = 0; ct < 4; ++ct) {               // 4 col tiles, reuse A
            const int col = ct * 16 + laneM;
            const v2f b = *(const v2f*)(sWp + bRow + col * 2);  // ds_load_b64
            acc[ct] = __builtin_amdgcn_wmma_f32_16x16x4_f32(
                false, a, false, b, (short)0, acc[ct], false, false);
        }
    }

    #pragma unroll
    for (int ct = 0; ct < 4; ++ct) {
        const int col = ct * 16 + laneM;
        const float bv = sB[col];
        #pragma unroll
        for (int v = 0; v < 8; ++v) {
            const int row = rowBase + v + halfSel * 8;
            if (row < nNodes) {
                const float val = acc[ct][v] + bv;
                if (mode == 0) {
                    H[(size_t)row * D_FEAT + col] = fmaxf(val, 0.f);
                } else {
                    unsafeAtomicAdd(&pool[(size_t)gid[row] * D_FEAT + col], val);
                }
            }
        }
    }
}

// ---------------------------------------------------------------------------
extern "C" void kernel_launch(void* const* d_in, const int* in_sizes, int n_in,
                              void* d_out, int out_size, void* d_ws, size_t ws_size,
                              hipStream_t stream)
{
    const float* feats = (const float*)d_in[0];
    const int*   src   = (const int*)  d_in[1];
    const int*   dst   = (const int*)  d_in[2];
    const int*   gid   = (const int*)  d_in[3];
    const float* W1    = (const float*)d_in[4];
    const float* b1    = (const float*)d_in[5];
    const float* W2    = (const float*)d_in[6];
    const float* b2    = (const float*)d_in[7];
    float* out = (float*)d_out;

    const int nNodes = in_sizes[0] / D_FEAT;
    const int nEdges = in_sizes[1];

    float* agg = (float*)d_ws;                          // [nNodes][64]
    float* h   = agg + (size_t)nNodes * D_FEAT;         // [nNodes][64]
    const size_t featBytes = (size_t)nNodes * D_FEAT * sizeof(float);

    const dim3 eBlk(256);
    const dim3 eGrd((nEdges + 15) / 16);
    const int rowTiles = (nNodes + 15) / 16;
    const dim3 gGrd((rowTiles + 7) / 8);

    // Layer 1: aggregate + GEMM(relu)
    hipMemsetAsync(agg, 0, featBytes, stream);
    hipMemsetAsync(out, 0, (size_t)out_size * sizeof(float), stream);
    gin_scatter_kernel<<<eGrd, eBlk, 0, stream>>>(feats, src, dst, agg, nEdges);
    gin_layer_kernel<<<gGrd, 256, 0, stream>>>(feats, agg, W1, b1,
                                               h, nullptr, nullptr, nNodes, 0);

    // Layer 2: aggregate + GEMM fused with graph sum-pooling
    hipMemsetAsync(agg, 0, featBytes, stream);
    gin_scatter_kernel<<<eGrd, eBlk, 0, stream>>>(h, src, dst, agg, nEdges);
    gin_layer_kernel<<<gGrd, 256, 0, stream>>>(h, agg, W2, b2,
                                               nullptr, gid, out, nNodes, 1);
}